// GraphSAGEModel_12249246728503
// MI455X (gfx1250) — compile-verified
//
#include <hip/hip_runtime.h>
#include <hip/hip_bf16.h>
#include <math.h>

typedef float v2f __attribute__((ext_vector_type(2)));
typedef float v8f __attribute__((ext_vector_type(8)));

#define HID 64

// ---------------- utility kernels ----------------

__global__ void fill_f32(float* __restrict__ p, float v, int n) {
    int i = blockIdx.x * blockDim.x + threadIdx.x;
    if (i < n) p[i] = v;
}

// deg[dst[e]] += 1
__global__ void deg_kernel(const int* __restrict__ dst, float* __restrict__ deg, int E) {
    int e = blockIdx.x * blockDim.x + threadIdx.x;
    if (e < E) atomicAdd(&deg[dst[e]], 1.0f);
}

// ---------------- WMMA f32 GEMM: Y[M x 64] = X[M x K] @ W[K x 64] ----------------
// One wave computes a 16x64 output tile with 4 accumulators of V_WMMA_F32_16X16X4_F32.
// Requires M % 16 == 0 (N_NODES = 100000 = 6250 * 16) and K % 4 == 0.
// blockDim.x = 64 (2 waves), dynamic LDS = K*64*4 bytes.
// W is staged into LDS K-pair-interleaved: lW[(k/2)*128 + col*2 + (k&1)],
// so each lane's B operand {W[k][col], W[k+1][col]} is one aligned ds_load_b64.
__global__ void gemm_wmma_f32(const float* __restrict__ X, const float* __restrict__ W,
                              float* __restrict__ Y, int K) {
    extern __shared__ float lW[];
    const int tid = threadIdx.x;
    // stage interleaved: entry i -> pair p = i>>6, col = i&63
    for (int i = tid; i < (K / 2) * HID; i += blockDim.x) {
        int p = i >> 6;
        int col = i & 63;
        float2 w2;
        w2.x = W[(2 * p) * HID + col];
        w2.y = W[(2 * p + 1) * HID + col];
        *(float2*)(lW + p * (2 * HID) + col * 2) = w2;
    }
    __syncthreads();

    const int lane = tid & 31;
    const int wave = tid >> 5;
    const int tile = blockIdx.x * 2 + wave;
    const int row0 = tile * 16;
    const int rsub = lane & 15;              // row (A) / col (B,C) within half-wave
    const int koff = (lane < 16) ? 0 : 2;    // K sub-offset held by this half-wave
    const int phalf = lane >> 4;             // pair-index offset for this half-wave
    const int mbase = (lane < 16) ? 0 : 8;   // C/D row base for this half-wave

    v8f acc0 = {}, acc1 = {}, acc2 = {}, acc3 = {};
    const float* xrow = X + (size_t)(row0 + rsub) * K;
    const float2* lW2 = (const float2*)lW;

    for (int k0 = 0; k0 < K; k0 += 4) {
        // A 16x4 f32: v0 = K(koff), v1 = K(koff+1)   (8B-aligned pair load)
        const float2 av = *(const float2*)(xrow + k0 + koff);
        v2f a; a.x = av.x; a.y = av.y;

        // B pairs: one ds_load_b64 per N-tile, offsets 0/128/256/384 bytes
        const float2* bp = lW2 + (size_t)(k0 / 2 + phalf) * HID + rsub;
        const float2 f0 = bp[0];
        const float2 f1 = bp[16];
        const float2 f2 = bp[32];
        const float2 f3 = bp[48];
        v2f b0; b0.x = f0.x; b0.y = f0.y;
        v2f b1; b1.x = f1.x; b1.y = f1.y;
        v2f b2; b2.x = f2.x; b2.y = f2.y;
        v2f b3; b3.x = f3.x; b3.y = f3.y;

        acc0 = __builtin_amdgcn_wmma_f32_16x16x4_f32(false, a, false, b0, (short)0, acc0, false, false);
        acc1 = __builtin_amdgcn_wmma_f32_16x16x4_f32(false, a, false, b1, (short)0, acc1, false, false);
        acc2 = __builtin_amdgcn_wmma_f32_16x16x4_f32(false, a, false, b2, (short)0, acc2, false, false);
        acc3 = __builtin_amdgcn_wmma_f32_16x16x4_f32(false, a, false, b3, (short)0, acc3, false, false);
    }

    float* yb = Y + (size_t)(row0 + mbase) * HID + rsub;
#pragma unroll
    for (int r = 0; r < 8; ++r) {
        yb[(size_t)r * HID + 0]  = acc0[r];
        yb[(size_t)r * HID + 16] = acc1[r];
        yb[(size_t)r * HID + 32] = acc2[r];
        yb[(size_t)r * HID + 48] = acc3[r];
    }
}

// ---------------- edge aggregation: Agg[dst] += Xa[src], 64 channels ----------------
// 16 threads per edge, float4 per thread (coalesced 256B gather per edge).
__global__ void edge_agg_kernel(const int* __restrict__ src, const int* __restrict__ dst,
                                const float* __restrict__ Xa, float* __restrict__ Agg, int E) {
    int t = blockIdx.x * blockDim.x + threadIdx.x;
    int e = t >> 4;
    if (e >= E) return;
    int c = (t & 15) << 2;
    const float4 v = *(const float4*)(Xa + (size_t)src[e] * HID + c);
    float* o = Agg + (size_t)dst[e] * HID + c;
    atomicAdd(o + 0, v.x);
    atomicAdd(o + 1, v.y);
    atomicAdd(o + 2, v.z);
    atomicAdd(o + 3, v.w);
}

// ---------------- combine: H = relu(Agg/max(deg,1) + Self + bias) ----------------
__global__ void combine_kernel(const float* __restrict__ Agg, const float* __restrict__ Self,
                               const float* __restrict__ deg, const float* __restrict__ bias,
                               float* __restrict__ H, int total) {
    int i = blockIdx.x * blockDim.x + threadIdx.x;
    if (i >= total) return;
    int node = i >> 6;
    int c = i & 63;
    float d = fmaxf(deg[node], 1.0f);
    float h = Agg[i] / d + Self[i] + bias[c];
    H[i] = fmaxf(h, 0.0f);
}

// ---------------- pooling ----------------
// h >= 0 (post-ReLU) so integer atomicMax on float bits == float max (init 0).
__global__ void pool_kernel(const float* __restrict__ H, const int* __restrict__ batch,
                            float* __restrict__ gsum, int* __restrict__ gmax, int total) {
    int i = blockIdx.x * blockDim.x + threadIdx.x;
    if (i >= total) return;
    int node = i >> 6;
    int c = i & 63;
    int b = batch[node];
    float h = H[i];
    atomicAdd(&gsum[b * HID + c], h);
    atomicMax(&gmax[b * HID + c], __float_as_int(h));
}

__global__ void cnt_kernel(const int* __restrict__ batch, float* __restrict__ cnt, int N) {
    int i = blockIdx.x * blockDim.x + threadIdx.x;
    if (i < N) atomicAdd(&cnt[batch[i]], 1.0f);
}

// ---------------- classifier: per-graph MLP ----------------
// block = graph, 64 threads; g = [mean(64), max(64)]; z = relu(g@W1+b1); out = sigmoid(z@W2+b2)
__global__ void classifier_kernel(const float* __restrict__ gsum, const int* __restrict__ gmax,
                                  const float* __restrict__ cnt,
                                  const float* __restrict__ w1, const float* __restrict__ b1,
                                  const float* __restrict__ w2, const float* __restrict__ b2,
                                  float* __restrict__ out) {
    __shared__ float red[HID];
    const int g = blockIdx.x;
    const int j = threadIdx.x;
    const float inv = 1.0f / fmaxf(cnt[g], 1.0f);
    float acc = b1[j];
#pragma unroll 4
    for (int k = 0; k < HID; ++k)
        acc += (gsum[g * HID + k] * inv) * w1[k * HID + j];
#pragma unroll 4
    for (int k = 0; k < HID; ++k)
        acc += __int_as_float(gmax[g * HID + k]) * w1[(HID + k) * HID + j];
    float z = fmaxf(acc, 0.0f);
    red[j] = z * w2[j];
    __syncthreads();
    for (int s = HID / 2; s > 0; s >>= 1) {
        if (j < s) red[j] += red[j + s];
        __syncthreads();
    }
    if (j == 0) {
        float logit = red[0] + b2[0];
        out[g] = 1.0f / (1.0f + expf(-logit));
    }
}

// ---------------- launcher ----------------

static inline int cdiv(long a, int b) { return (int)((a + b - 1) / b); }

extern "C" void kernel_launch(void* const* d_in, const int* in_sizes, int n_in,
                              void* d_out, int out_size, void* d_ws, size_t ws_size,
                              hipStream_t stream) {
    const float* x     = (const float*)d_in[0];
    const int*   ei    = (const int*)d_in[1];
    const int*   batch = (const int*)d_in[2];
    const float* l0_wl = (const float*)d_in[3];
    const float* l0_bl = (const float*)d_in[4];
    const float* l0_wr = (const float*)d_in[5];
    const float* l1_wl = (const float*)d_in[6];
    const float* l1_bl = (const float*)d_in[7];
    const float* l1_wr = (const float*)d_in[8];
    const float* w1    = (const float*)d_in[9];
    const float* b1    = (const float*)d_in[10];
    const float* w2    = (const float*)d_in[11];
    const float* b2    = (const float*)d_in[12];

    const int N  = in_sizes[2];          // 100000 (multiple of 16)
    const int E  = in_sizes[1] / 2;      // 3200000
    const int IN = in_sizes[0] / N;      // 128
    const int G  = out_size;             // 256
    const int NH = N * HID;

    float* ws   = (float*)d_ws;
    float* deg  = ws;                    // N (padded to 100352)
    float* A    = ws + 100352;           // N*64  (projected-for-aggregation)
    float* B    = A + (size_t)NH;        // N*64  (self path)
    float* C    = B + (size_t)NH;        // N*64  (aggregation accumulator)
    float* H    = C + (size_t)NH;        // N*64  (hidden activations)
    float* gsum = H + (size_t)NH;        // G*64
    int*   gmax = (int*)(gsum + G * HID);// G*64
    float* cnt  = (float*)(gmax + G * HID); // G

    const int TB = 256;
    const int gemm_blocks = (N / 16) / 2;        // 2 waves per block, 1 tile each

    // init accumulators
    fill_f32<<<cdiv(N, TB), TB, 0, stream>>>(deg, 0.0f, N);
    fill_f32<<<cdiv(NH, TB), TB, 0, stream>>>(C, 0.0f, NH);
    fill_f32<<<cdiv(G * HID, TB), TB, 0, stream>>>(gsum, 0.0f, G * HID);
    fill_f32<<<cdiv(G * HID, TB), TB, 0, stream>>>((float*)gmax, 0.0f, G * HID);
    fill_f32<<<cdiv(G, TB), TB, 0, stream>>>(cnt, 0.0f, G);

    const int* src = ei;
    const int* dstp = ei + E;

    deg_kernel<<<cdiv(E, TB), TB, 0, stream>>>(dstp, deg, E);

    // ---- layer 0: project first (128->64), then aggregate 64 channels ----
    gemm_wmma_f32<<<gemm_blocks, 64, IN * HID * sizeof(float), stream>>>(x, l0_wl, A, IN);
    gemm_wmma_f32<<<gemm_blocks, 64, IN * HID * sizeof(float), stream>>>(x, l0_wr, B, IN);
    edge_agg_kernel<<<cdiv((long)E * 16, TB), TB, 0, stream>>>(src, dstp, A, C, E);
    combine_kernel<<<cdiv(NH, TB), TB, 0, stream>>>(C, B, deg, l0_bl, H, NH);

    // ---- layer 1 ----
    fill_f32<<<cdiv(NH, TB), TB, 0, stream>>>(C, 0.0f, NH);
    gemm_wmma_f32<<<gemm_blocks, 64, HID * HID * sizeof(float), stream>>>(H, l1_wl, A, HID);
    gemm_wmma_f32<<<gemm_blocks, 64, HID * HID * sizeof(float), stream>>>(H, l1_wr, B, HID);
    edge_agg_kernel<<<cdiv((long)E * 16, TB), TB, 0, stream>>>(src, dstp, A, C, E);
    combine_kernel<<<cdiv(NH, TB), TB, 0, stream>>>(C, B, deg, l1_bl, H, NH);

    // ---- pooling + classifier ----
    pool_kernel<<<cdiv(NH, TB), TB, 0, stream>>>(H, batch, gsum, gmax, NH);
    cnt_kernel<<<cdiv(N, TB), TB, 0, stream>>>(batch, cnt, N);
    classifier_kernel<<<G, HID, 0, stream>>>(gsum, gmax, cnt, w1, b1, w2, b2, (float*)d_out);
}